// Kitsune_25812753449645
// MI455X (gfx1250) — compile-verified
//
#include <hip/hip_runtime.h>

typedef __attribute__((ext_vector_type(2))) float v2f;
typedef __attribute__((ext_vector_type(8))) float v8f;
typedef __attribute__((ext_vector_type(4))) unsigned int u32x4;
typedef __attribute__((ext_vector_type(8))) int i32x8;
typedef __attribute__((ext_vector_type(4))) int i32x4;

#define WAVES_PER_BLOCK 2
#define THREADS (WAVES_PER_BLOCK * 32)

#if defined(__gfx1250__) && __has_builtin(__builtin_amdgcn_tensor_load_to_lds) && \
    __has_builtin(__builtin_amdgcn_s_wait_tensorcnt)
#define USE_TDM 1
#else
#define USE_TDM 0
#endif

__device__ __forceinline__ float fast_sigmoid(float v) {
    // sigmoid(v) = 1 / (1 + exp(-v)) via native v_exp_f32 (exp2) + v_rcp_f32
    float e = __builtin_amdgcn_exp2f(v * -1.44269504088896340736f);
    return __builtin_amdgcn_rcpf(1.0f + e);
}

#if USE_TDM
// Issue one TDM DMA: 6400 contiguous bytes (16 rows x 100 f32) -> LDS.
// D# per CDNA5 ISA 8.3/8.4: 1-D tile, data_size=4B, tile_dim0=tensor_dim0=1600.
// This toolchain's builtin takes 6 args: (g0 x4, g1 x8, g2 x4, g3 x4, extra x8, cpol).
__device__ __forceinline__ void tdm_load_tile(const float* gsrc, unsigned ldsOff) {
    unsigned long long ga = (unsigned long long)(uintptr_t)gsrc;
    u32x4 g0;
    g0.x = 1u;                                                 // count=1, no gather
    g0.y = ldsOff;                                             // lds_addr (bytes)
    g0.z = (unsigned)(ga & 0xFFFFFFFFu);                       // global_addr[31:0]
    g0.w = (unsigned)((ga >> 32) & 0x01FFFFFFu) | 0x80000000u; // global_addr[56:32] | type=2
    i32x8 g1;
    g1[0] = (int)(2u << 16);      // workgroup_mask=0, data_size=2 (4 bytes)
    g1[1] = (int)(1600u << 16);   // tensor_dim0[15:0] in bits[63:48]
    g1[2] = (int)(1u << 16);      // tensor_dim0[31:16]=0, tensor_dim1[15:0]=1
    g1[3] = (int)(1600u << 16);   // tensor_dim1[31:16]=0, tile_dim0=1600
    g1[4] = 0;                    // tile_dim1=0 (1-D), tile_dim2=0
    g1[5] = 1600;                 // tensor_dim0_stride[31:0]
    g1[6] = 0;
    g1[7] = 0;
    i32x4 z4 = {0, 0, 0, 0};      // groups 2/3 unused for <=2-D
    i32x8 z8 = {0, 0, 0, 0, 0, 0, 0, 0};
    __builtin_amdgcn_tensor_load_to_lds(g0, g1, z4, z4, z8, 0);
}
#endif

__global__ __launch_bounds__(THREADS)
__attribute__((amdgpu_waves_per_eu(4)))
void kitsune_kernel(
    const float* __restrict__ x,
    const float* __restrict__ enc_w, const float* __restrict__ enc_b,
    const float* __restrict__ dec_w, const float* __restrict__ dec_b,
    const float* __restrict__ t_nmin, const float* __restrict__ t_nmax,
    const float* __restrict__ h_enc_w, const float* __restrict__ h_enc_b,
    const float* __restrict__ h_dec_w, const float* __restrict__ h_dec_b,
    const float* __restrict__ h_nmin, const float* __restrict__ h_nmax,
    float* __restrict__ out, int B)
{
    // ---- block-shared constant tables (filled once, then __syncthreads) ----
    __shared__ float sScale[100], sOffs[100];        // per-feature 1/(max-min+eps), min
    __shared__ float sEncFrag[10 * 3 * 2 * 32];      // [c][kstep][j][lane] B-fragments (enc)
    __shared__ float sDecFrag[10 * 2 * 2 * 32];      // [c][kstep][j][lane] B-fragments (dec)
    __shared__ float sEncB[80], sDecB[100];
    __shared__ float sHScale[10], sHOffs[10];
    __shared__ float sHeW[80], sHeB[8], sHdW[80], sHdB[10];
    // ---- per-wave working buffers (x tile double-buffered for TDM pipeline) ----
    __shared__ __align__(16) float sX[WAVES_PER_BLOCK][2][16 * 100];
    __shared__ float sH[WAVES_PER_BLOCK][16 * 9];    // enc output, padded stride 9
    __shared__ float sT[WAVES_PER_BLOCK][16 * 10];   // normalized tail scores

    const int tid = threadIdx.x;

    for (int i = tid; i < 100; i += THREADS) {
        float mn = t_nmin[i];
        sOffs[i]  = mn;
        sScale[i] = __builtin_amdgcn_rcpf(t_nmax[i] - mn + 1e-16f);
    }
    // Encoder B-fragments: B[k][n] = enc_w[c][n][k], n<8, k<10 (K padded to 12)
    // Fragment layout (4x16 f32 B): vgpr j, lane l -> K = kstep*4 + 2*(l/16) + j, N = l%16
    for (int i = tid; i < 10 * 3 * 2 * 32; i += THREADS) {
        int lane = i & 31; int r = i >> 5;
        int j = r & 1; r >>= 1;
        int s = r % 3; int c = r / 3;
        int N = lane & 15;
        int K = s * 4 + ((lane >> 4) << 1) + j;
        float v = 0.0f;
        if (N < 8 && K < 10) v = enc_w[c * 80 + N * 10 + K];
        sEncFrag[i] = v;
    }
    // Decoder B-fragments: B[k][n] = dec_w[c][n][k], n<10, k<8 (K exactly 8)
    for (int i = tid; i < 10 * 2 * 2 * 32; i += THREADS) {
        int lane = i & 31; int r = i >> 5;
        int j = r & 1; r >>= 1;
        int s = r & 1; int c = r >> 1;
        int N = lane & 15;
        int K = s * 4 + ((lane >> 4) << 1) + j;
        float v = 0.0f;
        if (N < 10) v = dec_w[c * 80 + N * 8 + K];
        sDecFrag[i] = v;
    }
    for (int i = tid; i < 80;  i += THREADS) sEncB[i] = enc_b[i];
    for (int i = tid; i < 100; i += THREADS) sDecB[i] = dec_b[i];
    for (int i = tid; i < 10;  i += THREADS) {
        float mn = h_nmin[i];
        sHOffs[i]  = mn;
        sHScale[i] = __builtin_amdgcn_rcpf(h_nmax[i] - mn + 1e-16f);
        sHdB[i]    = h_dec_b[i];
    }
    for (int i = tid; i < 80; i += THREADS) { sHeW[i] = h_enc_w[i]; sHdW[i] = h_dec_w[i]; }
    for (int i = tid; i < 8;  i += THREADS) sHeB[i] = h_enc_b[i];
    __syncthreads();

    const int lane  = tid & 31;
    const int wave  = tid >> 5;
    const int laneN = lane & 15;   // N / row-within-half index
    const int laneH = lane >> 4;   // half-wave select
    float* xBuf0 = sX[wave][0];
    float* xBuf1 = sX[wave][1];
    float* hT = sH[wave];
    float* tT = sT[wave];

    const int nTiles     = B >> 4;
    const int waveGlobal = blockIdx.x * WAVES_PER_BLOCK + wave;
    const int nWaves     = gridDim.x * WAVES_PER_BLOCK;
    float* outT = out + (size_t)B * 10;

#if USE_TDM
    // LDS byte offsets: flat LDS pointers carry the byte offset in addr[31:0]
    const unsigned ldsOff0 = (unsigned)(uintptr_t)xBuf0;
    const unsigned ldsOff1 = (unsigned)(uintptr_t)xBuf1;
    // pipeline prologue: DMA first tile into buffer 0
    if (waveGlobal < nTiles)
        tdm_load_tile(x + (size_t)waveGlobal * 1600, ldsOff0);
#endif

    int cur = 0;
    #pragma clang loop unroll(disable)
    for (int tile = waveGlobal; tile < nTiles; tile += nWaves) {
        float* xT;
#if USE_TDM
        xT = cur ? xBuf1 : xBuf0;
        {
            int nxt = tile + nWaves;
            if (nxt < nTiles) {
                // all LDS reads of the other buffer (previous iteration) retired
                asm volatile("s_wait_dscnt 0x0" ::: "memory");
                tdm_load_tile(x + (size_t)nxt * 1600, cur ? ldsOff0 : ldsOff1);
                __builtin_amdgcn_s_wait_tensorcnt(1);  // oldest DMA (this tile) done
            } else {
                __builtin_amdgcn_s_wait_tensorcnt(0);
            }
        }
#else
        xT = xBuf0;
        {   // ---- manual staged copy fallback ----
            const float4* src = (const float4*)(x + (size_t)tile * 1600);
            float4* dst = (float4*)xT;
            #pragma unroll
            for (int i = 0; i < 12; ++i) dst[lane + i * 32] = src[lane + i * 32];
            if (lane < 16) dst[lane + 384] = src[lane + 384];
            int nxt = tile + nWaves;
            if (nxt < nTiles) {
                const char* p = (const char*)(x + (size_t)nxt * 1600);
                __builtin_prefetch(p + lane * 200, 0, 0);
            }
        }
#endif

        // NOTE: keep this loop rolled -- full unroll blows past 256 VGPRs and
        // triggers s_set_vgpr_msb churn + occupancy collapse.
        #pragma clang loop unroll(disable)
        for (int c = 0; c < 10; ++c) {
            const int cb = c * 10;
            // ---- tail encoder: [16x10] @ [10x8] via 3x v_wmma_f32_16x16x4_f32 ----
            v8f acc = {0,0,0,0,0,0,0,0};
            #pragma unroll
            for (int s = 0; s < 3; ++s) {
                int K0 = s * 4 + (laneH << 1);
                int k0 = (K0     < 10) ? K0     : 0;   // clamp for safe LDS read
                int k1 = (K0 + 1 < 10) ? K0 + 1 : 0;
                float x0 = (xT[laneN * 100 + cb + k0] - sOffs[cb + k0]) * sScale[cb + k0];
                float x1 = (xT[laneN * 100 + cb + k1] - sOffs[cb + k1]) * sScale[cb + k1];
                v2f a;
                a.x = (K0     < 10) ? x0 : 0.0f;       // zero-pad K=10,11
                a.y = (K0 + 1 < 10) ? x1 : 0.0f;
                const float* bf = &sEncFrag[((c * 3 + s) * 2) * 32 + lane];
                v2f b; b.x = bf[0]; b.y = bf[32];
                acc = __builtin_amdgcn_wmma_f32_16x16x4_f32(
                        false, a, false, b, (short)0, acc, false, false);
            }
            // h = sigmoid(acc + b_enc) ; C-layout -> A-layout transpose via LDS
            {
                float be = sEncB[c * 8 + (laneN & 7)];
                bool on = laneN < 8;
                #pragma unroll
                for (int i = 0; i < 8; ++i) {
                    float hv = fast_sigmoid(acc[i] + be);
                    if (on) hT[(i + (laneH << 3)) * 9 + laneN] = hv;
                }
            }
            // ---- tail decoder: [16x8] @ [8x10] via 2x v_wmma_f32_16x16x4_f32 ----
            v8f dacc = {0,0,0,0,0,0,0,0};
            #pragma unroll
            for (int s = 0; s < 2; ++s) {
                int K0 = s * 4 + (laneH << 1);
                v2f a; a.x = hT[laneN * 9 + K0]; a.y = hT[laneN * 9 + K0 + 1];
                const float* bf = &sDecFrag[((c * 2 + s) * 2) * 32 + lane];
                v2f b; b.x = bf[0]; b.y = bf[32];
                dacc = __builtin_amdgcn_wmma_f32_16x16x4_f32(
                        false, a, false, b, (short)0, dacc, false, false);
            }
            // rec = sigmoid(dacc + b_dec); squared error vs xn; reduce over features
            float rs[8];
            {
                int   nf  = (laneN < 10) ? laneN : 0;
                float bd  = sDecB[cb + nf];
                float off = sOffs[cb + nf];
                float scl = sScale[cb + nf];
                #pragma unroll
                for (int i = 0; i < 8; ++i) {
                    int row = i + (laneH << 3);
                    float rec = fast_sigmoid(dacc[i] + bd);
                    float xn  = (xT[row * 100 + cb + nf] - off) * scl;
                    float d   = rec - xn;
                    rs[i] = (laneN < 10) ? d * d : 0.0f;
                }
            }
            // butterfly-sum across the 16 lanes of each half (wave32)
            #pragma unroll
            for (int m = 1; m <= 8; m <<= 1) {
                #pragma unroll
                for (int i = 0; i < 8; ++i) rs[i] += __shfl_xor(rs[i], m, 32);
            }
            if (laneN == 0) {
                float hs = sHScale[c], ho = sHOffs[c];
                #pragma unroll
                for (int i = 0; i < 8; ++i) {
                    float loss = __builtin_sqrtf(rs[i] * 0.1f);
                    loss = (loss == 0.0f) ? 0.01f : loss;
                    tT[(i + (laneH << 3)) * 10 + c] = (loss - ho) * hs;
                }
            }
        }

        // ---- head autoencoder (10 -> 8 -> 10), one row per lane ----
        const int row = tile * 16 + laneN;
        if (laneH == 0) {
            float tv[10];
            #pragma unroll
            for (int c2 = 0; c2 < 10; ++c2) tv[c2] = tT[laneN * 10 + c2];
            float hh[8];
            #pragma unroll
            for (int j = 0; j < 8; ++j) {
                float s = sHeB[j];
                #pragma unroll
                for (int c2 = 0; c2 < 10; ++c2) s += tv[c2] * sHeW[j * 10 + c2];
                hh[j] = fast_sigmoid(s);
            }
            float* o = out + (size_t)row * 10;
            #pragma unroll
            for (int f = 0; f < 10; ++f) {
                float s = sHdB[f];
                #pragma unroll
                for (int j = 0; j < 8; ++j) s += hh[j] * sHdW[f * 8 + j];
                o[f] = fast_sigmoid(s);
            }
        } else {
            // lanes 16..31 write out the normalized tail vector t
            float* o = outT + (size_t)row * 10;
            #pragma unroll
            for (int c2 = 0; c2 < 10; ++c2) o[c2] = tT[laneN * 10 + c2];
        }
        cur ^= 1;
    }
}

extern "C" void kernel_launch(void* const* d_in, const int* in_sizes, int n_in,
                              void* d_out, int out_size, void* d_ws, size_t ws_size,
                              hipStream_t stream) {
    (void)n_in; (void)out_size; (void)d_ws; (void)ws_size;
    const float* x    = (const float*)d_in[0];
    const float* encw = (const float*)d_in[1];
    const float* encb = (const float*)d_in[2];
    const float* decw = (const float*)d_in[3];
    const float* decb = (const float*)d_in[4];
    const float* nmin = (const float*)d_in[5];
    const float* nmax = (const float*)d_in[6];
    const float* hew  = (const float*)d_in[7];
    const float* heb  = (const float*)d_in[8];
    const float* hdw  = (const float*)d_in[9];
    const float* hdb  = (const float*)d_in[10];
    const float* hmin = (const float*)d_in[11];
    const float* hmax = (const float*)d_in[12];
    float* out = (float*)d_out;

    int B = in_sizes[0] / 100;
    int nTiles = B >> 4;
    int blocks = (nTiles + WAVES_PER_BLOCK - 1) / WAVES_PER_BLOCK;
    if (blocks > 4096) blocks = 4096;   // ~4 tiles per wave -> pipeline depth
    if (blocks < 1) blocks = 1;

    hipLaunchKernelGGL(kitsune_kernel, dim3(blocks), dim3(THREADS), 0, stream,
                       x, encw, encb, decw, decb, nmin, nmax,
                       hew, heb, hdw, hdb, hmin, hmax, out, B);
}